// FarthestPointSample_9698036154797
// MI455X (gfx1250) — compile-verified
//
#include <hip/hip_runtime.h>
#include <hip/hip_bf16.h>

// Farthest Point Sampling for MI455X (gfx1250).
// Shapes fixed by the reference: B=64, N=131072, C=3, npoint=4096.
//
// One 1024-thread workgroup (32 wave32 waves) per batch — exactly fills a
// WGP's wave slots, so LDS is free to take. Persistent 4096-iteration loop.
// Working set (96MB xyz + global dist remainder) is L2-resident (192MB L2).
// 59% of the per-batch distance array (77824 points, 304KB) lives
// permanently in LDS (320KB/WGP on CDNA5), cutting per-iteration global
// traffic to ~1.9MB/batch and moving most of the dist read-modify-write
// onto the LDS path (ds_load_b128/ds_store_b128).

#define BATCHES   64
#define NPTS      131072
#define NPOINT    4096
#define THREADS   1024
#define WAVES     (THREADS / 32)
#define CHUNKS    (NPTS / 4)          // 4 points per chunk (48B xyz, 16B dist)
#define K_PER_T   (CHUNKS / THREADS)  // 32 chunks per thread
#define K_LDS     19                  // chunks k<19 keep dist in LDS (77824 pts)
#define LDS_CHUNKS (K_LDS * THREADS)  // 19456 float4 = 304 KB

__global__ __launch_bounds__(THREADS, 1)
void FarthestPointSample_9698036154797_kernel(const float* __restrict__ xyz,
                                              float* __restrict__ dist,
                                              int* __restrict__ out)
{
    const int b    = blockIdx.x;
    const int tid  = threadIdx.x;
    const int lane = tid & 31;
    const int wave = tid >> 5;

    const float*  xyzB  = xyz + (size_t)b * (size_t)NPTS * 3u;
    const float4* xyz4  = reinterpret_cast<const float4*>(xyzB);
    float4*       distB = reinterpret_cast<float4*>(dist + (size_t)b * (size_t)NPTS);
    int*          outB  = out + (size_t)b * (size_t)NPOINT;

    __shared__ float4 distLds4[LDS_CHUNKS];   // 304 KB: dist for points 0..77823
    __shared__ float  sD[WAVES];
    __shared__ int    sI[WAVES];
    __shared__ int    sWin;

    // Initialize min-distance to 1e10 (graph-safe, done every call).
    const float4 big = make_float4(1e10f, 1e10f, 1e10f, 1e10f);
    #pragma unroll
    for (int k = 0; k < K_LDS; ++k)
        distLds4[tid + k * THREADS] = big;
    #pragma unroll
    for (int k = K_LDS; k < K_PER_T; ++k)
        distB[tid + k * THREADS] = big;
    __syncthreads();

    int farthest = 0;  // deterministic seed (reference seeds via JAX PRNG)

    for (int i = 0; i < NPOINT; ++i) {
        // Emit index BEFORE update (matches the reference scan).
        if (tid == 0) outB[i] = farthest;

        // Centroid gather: force uniform -> scalar/broadcast load from L2.
        const int fi = __builtin_amdgcn_readfirstlane(farthest);
        const float cx = xyzB[3 * (size_t)fi + 0];
        const float cy = xyzB[3 * (size_t)fi + 1];
        const float cz = xyzB[3 * (size_t)fi + 2];

        float bestD = -1.0f;
        int   bestI = 0;

        // ---- Region 1: dist resident in LDS (chunks k = 0..18) ----
        #pragma unroll 4
        for (int k = 0; k < K_LDS; ++k) {
            const int c = tid + k * THREADS;  // thread-strided: wave-contiguous VMEM
            __builtin_prefetch((const void*)&xyz4[3 * (c + THREADS)], 0, 3);

            const float4 q0 = xyz4[3 * c + 0];  // x0 y0 z0 x1
            const float4 q1 = xyz4[3 * c + 1];  // y1 z1 x2 y2
            const float4 q2 = xyz4[3 * c + 2];  // z2 x3 y3 z3
            float4 dv = distLds4[c];

            float dx, dy, dz, d;
            dx = q0.x - cx; dy = q0.y - cy; dz = q0.z - cz;
            d = dx * dx + dy * dy + dz * dz; dv.x = fminf(dv.x, d);
            dx = q0.w - cx; dy = q1.x - cy; dz = q1.y - cz;
            d = dx * dx + dy * dy + dz * dz; dv.y = fminf(dv.y, d);
            dx = q1.z - cx; dy = q1.w - cy; dz = q2.x - cz;
            d = dx * dx + dy * dy + dz * dz; dv.z = fminf(dv.z, d);
            dx = q2.y - cx; dy = q2.z - cy; dz = q2.w - cz;
            d = dx * dx + dy * dy + dz * dz; dv.w = fminf(dv.w, d);

            distLds4[c] = dv;

            const int p = 4 * c;
            if (dv.x > bestD || (dv.x == bestD && p     < bestI)) { bestD = dv.x; bestI = p;     }
            if (dv.y > bestD || (dv.y == bestD && p + 1 < bestI)) { bestD = dv.y; bestI = p + 1; }
            if (dv.z > bestD || (dv.z == bestD && p + 2 < bestI)) { bestD = dv.z; bestI = p + 2; }
            if (dv.w > bestD || (dv.w == bestD && p + 3 < bestI)) { bestD = dv.w; bestI = p + 3; }
        }

        // ---- Region 2: dist resident in global (L2) (chunks k = 19..31) ----
        #pragma unroll 4
        for (int k = K_LDS; k < K_PER_T; ++k) {
            const int c = tid + k * THREADS;
            if (k + 1 < K_PER_T)
                __builtin_prefetch((const void*)&xyz4[3 * (c + THREADS)], 0, 3);

            const float4 q0 = xyz4[3 * c + 0];
            const float4 q1 = xyz4[3 * c + 1];
            const float4 q2 = xyz4[3 * c + 2];
            float4 dv = distB[c];

            float dx, dy, dz, d;
            dx = q0.x - cx; dy = q0.y - cy; dz = q0.z - cz;
            d = dx * dx + dy * dy + dz * dz; dv.x = fminf(dv.x, d);
            dx = q0.w - cx; dy = q1.x - cy; dz = q1.y - cz;
            d = dx * dx + dy * dy + dz * dz; dv.y = fminf(dv.y, d);
            dx = q1.z - cx; dy = q1.w - cy; dz = q2.x - cz;
            d = dx * dx + dy * dy + dz * dz; dv.z = fminf(dv.z, d);
            dx = q2.y - cx; dy = q2.z - cy; dz = q2.w - cz;
            d = dx * dx + dy * dy + dz * dz; dv.w = fminf(dv.w, d);

            distB[c] = dv;

            const int p = 4 * c;
            if (dv.x > bestD || (dv.x == bestD && p     < bestI)) { bestD = dv.x; bestI = p;     }
            if (dv.y > bestD || (dv.y == bestD && p + 1 < bestI)) { bestD = dv.y; bestI = p + 1; }
            if (dv.z > bestD || (dv.z == bestD && p + 2 < bestI)) { bestD = dv.z; bestI = p + 2; }
            if (dv.w > bestD || (dv.w == bestD && p + 3 < bestI)) { bestD = dv.w; bestI = p + 3; }
        }

        // Wave32 butterfly reduction (ds_bpermute path on gfx1250).
        #pragma unroll
        for (int off = 16; off > 0; off >>= 1) {
            float oD = __shfl_xor(bestD, off, 32);
            int   oI = __shfl_xor(bestI, off, 32);
            if (oD > bestD || (oD == bestD && oI < bestI)) { bestD = oD; bestI = oI; }
        }
        if (lane == 0) { sD[wave] = bestD; sI[wave] = bestI; }
        __syncthreads();

        if (wave == 0) {
            float d2 = sD[lane];   // WAVES == 32 exactly
            int   i2 = sI[lane];
            #pragma unroll
            for (int off = 16; off > 0; off >>= 1) {
                float oD = __shfl_xor(d2, off, 32);
                int   oI = __shfl_xor(i2, off, 32);
                if (oD > d2 || (oD == d2 && oI < i2)) { d2 = oD; i2 = oI; }
            }
            if (lane == 0) sWin = i2;
        }
        __syncthreads();
        farthest = sWin;
        // Safe without a third barrier: next iteration's sD/sI writes happen
        // only after every wave passed the barrier above.
    }
}

extern "C" void kernel_launch(void* const* d_in, const int* in_sizes, int n_in,
                              void* d_out, int out_size, void* d_ws, size_t ws_size,
                              hipStream_t stream) {
    (void)in_sizes; (void)n_in; (void)out_size; (void)ws_size;
    const float* xyz  = (const float*)d_in[0];   // [64, 131072, 3] f32
    float*       dist = (float*)d_ws;            // [64, 131072] f32 scratch (only k>=19 region used per batch)
    int*         out  = (int*)d_out;             // [64, 4096] int32

    FarthestPointSample_9698036154797_kernel<<<BATCHES, THREADS, 0, stream>>>(xyz, dist, out);
}